// DefaultMoeRoutingMethod_45303315038675
// MI455X (gfx1250) — compile-verified
//
#include <hip/hip_runtime.h>
#include <stdint.h>

#define NUM_EXPERTS        64
#define TOP_K_FIXED        2
#define WAVE_SIZE          32
#define WAVES_PER_BLOCK    4
#define THREADS_PER_BLOCK  (WAVES_PER_BLOCK * WAVE_SIZE)   // 128
#define TOKENS_PER_WAVE    32
#define TOKENS_PER_BLOCK   (WAVES_PER_BLOCK * TOKENS_PER_WAVE) // 128
#define ROW_BYTES          (NUM_EXPERTS * 4)               // 256
#define ROW_PAD_BYTES      (ROW_BYTES + 16)                // 272: kills LDS bank conflicts
#define LDS_BYTES_PER_WAVE (TOKENS_PER_WAVE * ROW_PAD_BYTES) // 8704

typedef float v4f __attribute__((ext_vector_type(4)));

// Scan 64 logits: top-2 (monotonic under softmax), then softmax denominator.
// Sequential '>' scan exactly matches jax.lax.top_k tie-breaking
// (first occurrence wins -> ascending indices on exact-fp32 ties).
__device__ __forceinline__ void route_token(const v4f* __restrict__ row,
                                            int token,
                                            int* __restrict__ idx_out,
                                            float* __restrict__ val_out) {
  v4f r[NUM_EXPERTS / 4];
#pragma unroll
  for (int i = 0; i < NUM_EXPERTS / 4; ++i) r[i] = row[i];

  float b1 = -3.402823466e38f, b2 = -3.402823466e38f;
  int   i1 = 0, i2 = 0;
#pragma unroll
  for (int i = 0; i < NUM_EXPERTS / 4; ++i) {
#pragma unroll
    for (int c = 0; c < 4; ++c) {
      float x  = r[i][c];
      int   e  = i * 4 + c;
      bool gt1 = x > b1;
      bool gt2 = x > b2;
      float nb2 = gt1 ? b1 : (gt2 ? x : b2);
      int   ni2 = gt1 ? i1 : (gt2 ? e : i2);
      b1 = gt1 ? x : b1;
      i1 = gt1 ? e : i1;
      b2 = nb2;
      i2 = ni2;
    }
  }

  // exp(x - b1) = exp2(x*log2e - b1*log2e): one v_fma per element.
  const float LOG2E = 1.4426950408889634f;
  const float nmax  = -b1 * LOG2E;
  float s0 = 0.0f, s1 = 0.0f, s2 = 0.0f, s3 = 0.0f;  // 4 independent chains
#pragma unroll
  for (int i = 0; i < NUM_EXPERTS / 4; ++i) {
    s0 += __builtin_amdgcn_exp2f(__builtin_fmaf(r[i][0], LOG2E, nmax));
    s1 += __builtin_amdgcn_exp2f(__builtin_fmaf(r[i][1], LOG2E, nmax));
    s2 += __builtin_amdgcn_exp2f(__builtin_fmaf(r[i][2], LOG2E, nmax));
    s3 += __builtin_amdgcn_exp2f(__builtin_fmaf(r[i][3], LOG2E, nmax));
  }
  float s   = (s0 + s1) + (s2 + s3);
  float inv = 1.0f / s;
  float p1  = inv;                                             // exp(b1-b1)/s
  float p2  = __builtin_amdgcn_exp2f(__builtin_fmaf(b2, LOG2E, nmax)) * inv;

  reinterpret_cast<int2*>(idx_out)[token]   = make_int2(i1, i2);     // 8B coalesced
  reinterpret_cast<float2*>(val_out)[token] = make_float2(p1, p2);   // 8B coalesced
}

__global__ __launch_bounds__(THREADS_PER_BLOCK)
void moe_topk_softmax_kernel(const float* __restrict__ logits,
                             int* __restrict__ idx_out,
                             float* __restrict__ val_out,
                             int num_tokens) {
  __shared__ __align__(16) unsigned char lds_raw[WAVES_PER_BLOCK * LDS_BYTES_PER_WAVE];

  const int lane       = threadIdx.x & (WAVE_SIZE - 1);
  const int wave       = threadIdx.x >> 5;
  const int token_base = (blockIdx.x * WAVES_PER_BLOCK + wave) * TOKENS_PER_WAVE;
  if (token_base >= num_tokens) return;  // wave-uniform

  if (token_base + TOKENS_PER_WAVE <= num_tokens) {
    // ---- Fast path: async global->LDS stage of 32 rows (8 KB) per wave ----
    // LDS byte address: low 32 bits of the generic pointer are the wave-relative
    // LDS address (gfx1250 aperture keeps the LDS offset in addr[31:0]).
    unsigned lds_wave = (unsigned)(uintptr_t)(&lds_raw[0]) +
                        (unsigned)(wave * LDS_BYTES_PER_WAVE);
    unsigned gbase = (unsigned)token_base * (unsigned)ROW_BYTES;  // < 2^28, fits i32

    // 16 iterations x (32 lanes x 16B) = 8192 B, fully coalesced 512B/wave bursts.
#pragma unroll
    for (int i = 0; i < (TOKENS_PER_WAVE * ROW_BYTES) / (WAVE_SIZE * 16); ++i) {
      unsigned goff = (unsigned)(i * (WAVE_SIZE * 16)) + (unsigned)(lane * 16);
      unsigned row  = goff >> 8;              // /256: source row within the stage
      unsigned col  = goff & (ROW_BYTES - 1); // byte within row
      unsigned ldst = lds_wave + row * ROW_PAD_BYTES + col;  // padded dest
      unsigned gvo  = gbase + goff;
      // GVS mode: reads from the SGPR base plus the VGPR offset, writes to the LDS address in vdst
      asm volatile("global_load_async_to_lds_b128 %0, %1, %2"
                   :
                   : "v"(ldst), "v"(gvo), "s"(logits)
                   : "memory");
    }
    // Wave-local completion: this wave only reads rows it staged itself.
    asm volatile("s_wait_asynccnt 0" ::: "memory");

    const v4f* lrow = reinterpret_cast<const v4f*>(
        lds_raw + wave * LDS_BYTES_PER_WAVE + lane * ROW_PAD_BYTES); // ds_load_b128
    route_token(lrow, token_base + lane, idx_out, val_out);
  } else {
    // ---- Tail path: direct global b128 loads, one token per lane ----
    int token = token_base + lane;
    if (token < num_tokens) {
      const v4f* grow =
          reinterpret_cast<const v4f*>(logits + (size_t)token * NUM_EXPERTS);
      route_token(grow, token, idx_out, val_out);
    }
  }
}

extern "C" void kernel_launch(void* const* d_in, const int* in_sizes, int n_in,
                              void* d_out, int out_size, void* d_ws, size_t ws_size,
                              hipStream_t stream) {
  (void)n_in; (void)out_size; (void)d_ws; (void)ws_size;
  const float* logits = (const float*)d_in[0];
  // d_in[1] is top_k (==2); fixed at compile time.
  int num_tokens = in_sizes[0] / NUM_EXPERTS;

  // d_out layout (tuple concat): [T*2] int32 indices, then [T*2] f32 values.
  int*   idx_out = (int*)d_out;
  float* val_out = (float*)d_out + (size_t)num_tokens * TOP_K_FIXED;

  int grid = (num_tokens + TOKENS_PER_BLOCK - 1) / TOKENS_PER_BLOCK;
  moe_topk_softmax_kernel<<<grid, THREADS_PER_BLOCK, 0, stream>>>(
      logits, idx_out, val_out, num_tokens);
}